// GroverFilteredAttention_11338713661467
// MI455X (gfx1250) — compile-verified
//
#include <hip/hip_runtime.h>
#include <hip/hip_bf16.h>
#include <math.h>

// ---------------------------------------------------------------------------
// GroverFilteredAttention for MI455X (gfx1250), fp32 WMMA path.
// B=4, S=1024, E=1024, H=16, Dh=64.  All matmuls via V_WMMA_F32_16X16X4_F32.
// GEMM A-tiles staged with GLOBAL_LOAD_ASYNC_TO_LDS_B128 (ASYNCcnt).
// ---------------------------------------------------------------------------

typedef float v2f __attribute__((ext_vector_type(2)));
typedef float v8f __attribute__((ext_vector_type(8)));

__device__ __forceinline__ v8f wmma4(v2f a, v2f b, v8f c) {
  // D = A(16x4) * B(4x16) + C(16x16), fp32.
  return __builtin_amdgcn_wmma_f32_16x16x4_f32(false, a, false, b, (short)0, c,
                                               false, false);
}

static __device__ __forceinline__ v8f v8f_zero() {
  v8f z = {0.f, 0.f, 0.f, 0.f, 0.f, 0.f, 0.f, 0.f};
  return z;
}

// ---------------------------------------------------------------------------
// Generic fp32 GEMM:  C[M,N] = A[M,K] @ W[K,N] + bias[N]
// 256 threads = 8 waves.  Block tile 64(M) x 128(N); waves arranged 4x2,
// each wave owns a 16x64 slab (4 WMMA accumulators sharing one A fragment).
//
// A tile (64x32, row-major, stride 36) is staged asynchronously with
// global_load_async_to_lds_b128; W tile is staged *transposed* (Wt[n][k],
// stride 38) so B fragments are contiguous 8B-aligned float2 -> ds_load_b64
// with zero register fix-up movs.  Requires K % 32 == 0 (true here).
// ---------------------------------------------------------------------------
__global__ __launch_bounds__(256) void gemm_f32_wmma(
    const float* __restrict__ A, const float* __restrict__ W,
    const float* __restrict__ bias, float* __restrict__ C,
    int M, int N, int K) {
  // Single shared object => LDS base offset 0 (needed: the async-to-LDS asm
  // takes raw per-lane LDS byte addresses as integers).
  __shared__ float smem[64 * 36 + 128 * 38];
  float(*As)[36] = reinterpret_cast<float(*)[36]>(smem);          // [64][36]
  float(*Wt)[38] = reinterpret_cast<float(*)[38]>(smem + 64 * 36); // [128][38]
  const unsigned WT_BASE = 64u * 36u * 4u;

  const int tid  = threadIdx.x;
  const int lane = tid & 31;
  const int wave = tid >> 5;
  const int wm   = wave & 3;        // 0..3 along M
  const int wn   = wave >> 2;       // 0..1 along N
  const int bm   = blockIdx.y * 64;
  const int bn   = blockIdx.x * 128;
  const int l15  = lane & 15;
  const int lh   = lane >> 4;       // 16-lane half
  const int koff = lh * 2;          // fragment K sub-offset

  // Async A staging: 64 rows x 8 chunks of 16B = 512 chunks, 2 per thread.
  // LDS addr (row*36 + kc)*4 is 16B aligned (144*row + 16*c).
  const int ar0 = (tid + 0)   >> 3, ac0 = ((tid + 0)   & 7) * 4;
  const int ar1 = (tid + 256) >> 3, ac1 = ((tid + 256) & 7) * 4;
  const unsigned alds0 = (unsigned)((ar0 * 36 + ac0) * 4);
  const unsigned alds1 = (unsigned)((ar1 * 36 + ac1) * 4);
  const unsigned agof0 = (unsigned)((ar0 * K + ac0) * 4);
  const unsigned agof1 = (unsigned)((ar1 * K + ac1) * 4);
  const float* Abase = A + (size_t)bm * K;  // uniform -> SGPR pair

  v8f acc[4];
#pragma unroll
  for (int t = 0; t < 4; ++t) acc[t] = v8f_zero();

  for (int k0 = 0; k0 < K; k0 += 32) {
    // ---- issue async A-tile copy (global -> LDS, ASYNCcnt) ----
    {
      unsigned g0 = agof0 + (unsigned)(k0 * 4);
      unsigned g1 = agof1 + (unsigned)(k0 * 4);
      asm volatile("global_load_async_to_lds_b128 %0, %1, %2"
                   :: "v"(alds0), "v"(g0), "s"(Abase) : "memory");
      asm volatile("global_load_async_to_lds_b128 %0, %1, %2"
                   :: "v"(alds1), "v"(g1), "s"(Abase) : "memory");
      (void)WT_BASE;
    }
    // ---- synchronous transposed W staging (overlaps the async copy) ----
#pragma unroll
    for (int i = 0; i < 16; ++i) {
      int idx = tid + i * 256;
      int n = idx & 127, kk = idx >> 7;          // coalesced 128B global reads
      Wt[n][kk] = W[(size_t)(k0 + kk) * N + (bn + n)];  // conflict-free stores
    }
    if (k0 + 32 < K)  // speculative prefetch of next W tile
      __builtin_prefetch(&W[(size_t)(k0 + 32 + (tid >> 7)) * N + bn + (tid & 127)], 0, 1);

    asm volatile("s_wait_asynccnt 0x0" ::: "memory");
    __syncthreads();

#pragma unroll
    for (int kk = 0; kk < 32; kk += 4) {
      v2f a = *reinterpret_cast<const v2f*>(&As[wm * 16 + l15][kk + koff]);
#pragma unroll
      for (int t = 0; t < 4; ++t) {
        v2f b = *reinterpret_cast<const v2f*>(&Wt[wn * 64 + t * 16 + l15][kk + koff]);
        acc[t] = wmma4(a, b, acc[t]);
      }
    }
    __syncthreads();
  }

  // Epilogue: bias + store.  C layout: vgpr i -> row i + 8*lh, col = l15.
#pragma unroll
  for (int t = 0; t < 4; ++t) {
    int n = bn + wn * 64 + t * 16 + l15;
    float bv = bias[n];
#pragma unroll
    for (int i = 0; i < 8; ++i) {
      int m = bm + wm * 16 + i + lh * 8;
      C[(size_t)m * N + n] = acc[t][i] + bv;
    }
  }
}

// ---------------------------------------------------------------------------
// Fused threshold attention.  Grid (8, H, B); 256 threads = 8 waves; each
// wave owns 16 query rows.  qkv is the raw [4096][3072] projection output:
//   q[b,h,s,d] = qkv[b*1024+s][      h*64+d]
//   k[b,h,s,d] = qkv[b*1024+s][1024 + h*64+d]
//   v[b,h,s,d] = qkv[b*1024+s][2048 + h*64+d]
// Writes: fa (unnormalized prob*keep), srow (row mask sums), attn_out rows.
// ---------------------------------------------------------------------------
__global__ __launch_bounds__(256) void attn_kernel(
    const float* __restrict__ qkv, float* __restrict__ fa,
    float* __restrict__ attn_out, float* __restrict__ srow) {
  const int b = blockIdx.z, h = blockIdx.y, qb = blockIdx.x;
  const int lane = threadIdx.x & 31;
  const int wave = threadIdx.x >> 5;
  const int q0   = qb * 128 + wave * 16;
  const int l15  = lane & 15;
  const int lh   = lane >> 4;
  const int koff = lh * 2;
  const float scale = 0.125f;  // 64^-0.5

  const float* Qp    = qkv + (size_t)(b * 1024 + q0 + l15) * 3072 + h * 64;
  const float* Kbase = qkv + (size_t)(b * 1024) * 3072 + 1024 + h * 64;
  const float* Vbase = qkv + (size_t)(b * 1024) * 3072 + 2048 + h * 64;

  // Preload all Q fragments for this wave's 16 rows (K=64 -> 16 steps of 4).
  v2f qa[16];
#pragma unroll
  for (int kb = 0; kb < 16; ++kb)
    qa[kb] = *reinterpret_cast<const v2f*>(&Qp[kb * 4 + koff]);

  // --- Pass 1: online softmax statistics (m = row max, l = sum exp) -------
  float m[8], l[8];
#pragma unroll
  for (int i = 0; i < 8; ++i) { m[i] = -1e30f; l[i] = 0.f; }

  for (int j0 = 0; j0 < 1024; j0 += 16) {
    v8f c = v8f_zero();
    const float* Kp = Kbase + (size_t)(j0 + l15) * 3072;
#pragma unroll
    for (int kb = 0; kb < 16; ++kb) {
      // B[kk][n] = K[j0+n][kk]: K row-major -> contiguous float2
      v2f bf = *reinterpret_cast<const v2f*>(&Kp[kb * 4 + koff]);
      c = wmma4(qa[kb], bf, c);
    }
#pragma unroll
    for (int i = 0; i < 8; ++i) {
      float v = c[i] * scale;
      if (v > m[i]) {
        l[i] = l[i] * __expf(m[i] - v) + 1.f;
        m[i] = v;
      } else {
        l[i] += __expf(v - m[i]);
      }
    }
  }
  // Cross-lane combine: each row's 16 columns live in one 16-lane half.
#pragma unroll
  for (int i = 0; i < 8; ++i) {
    float mi = m[i], li = l[i];
#pragma unroll
    for (int off = 1; off < 16; off <<= 1) {
      float m2 = __shfl_xor(mi, off, 32);
      float l2 = __shfl_xor(li, off, 32);
      float mn = fmaxf(mi, m2);
      li = li * __expf(mi - mn) + l2 * __expf(m2 - mn);
      mi = mn;
    }
    m[i] = mi; l[i] = li;
  }

  // any_selected  <=>  max prob = 1/l > 0.1  <=>  l < 10
  float invl[8]; bool anysel[8]; float s[8];
#pragma unroll
  for (int i = 0; i < 8; ++i) {
    invl[i] = 1.f / l[i];
    anysel[i] = (l[i] < 10.f);
    s[i] = 0.f;
  }

  v8f acc[4];
#pragma unroll
  for (int t = 0; t < 4; ++t) acc[t] = v8f_zero();

  // Per-wave LDS transpose buffer (pad 18: 8B-aligned float2 fragments).
  __shared__ float Pt[8][16][18];
  float(*P)[18] = Pt[wave];

  // --- Pass 2: recompute logits, mask, emit P, accumulate P@V -------------
  for (int j0 = 0; j0 < 1024; j0 += 16) {
    v8f c = v8f_zero();
    const float* Kp = Kbase + (size_t)(j0 + l15) * 3072;
#pragma unroll
    for (int kb = 0; kb < 16; ++kb) {
      v2f bf = *reinterpret_cast<const v2f*>(&Kp[kb * 4 + koff]);
      c = wmma4(qa[kb], bf, c);  // bit-identical to pass 1 -> v==m is exact
    }
#pragma unroll
    for (int i = 0; i < 8; ++i) {
      float v = c[i] * scale;
      float pm = __expf(v - m[i]);
      float prob = pm * invl[i];
      bool keep = anysel[i] ? (prob > 0.1f) : (v == m[i]);  // argmax fallback
      float t = keep ? prob : 0.f;
      s[i] += t;
      int row = i + lh * 8;
      fa[((size_t)((b * 16 + h) * 1024 + q0 + row)) * 1024 + j0 + l15] = t;
      P[row][l15] = t;  // intra-wave; LDS ops from one wave are in-order
    }
    // P@V: A = P(16x16), B = V tile (16 keys x 64 dims), 4 n-blocks.
#pragma unroll
    for (int kb = 0; kb < 4; ++kb) {
      v2f pa = *reinterpret_cast<const v2f*>(&P[l15][kb * 4 + koff]);
      const float* Vp0 = Vbase + (size_t)(j0 + kb * 4 + koff) * 3072;
      const float* Vp1 = Vp0 + 3072;
#pragma unroll
      for (int nb = 0; nb < 4; ++nb) {
        v2f bf;
        bf.x = Vp0[nb * 16 + l15];
        bf.y = Vp1[nb * 16 + l15];
        acc[nb] = wmma4(pa, bf, acc[nb]);
      }
    }
  }

  // Reduce masked sums across the 16-lane halves.
#pragma unroll
  for (int i = 0; i < 8; ++i) {
    float si = s[i];
#pragma unroll
    for (int off = 1; off < 16; off <<= 1) si += __shfl_xor(si, off, 32);
    s[i] = si;  // >= max prob > 0, never zero
  }
  if (l15 == 0) {
#pragma unroll
    for (int i = 0; i < 8; ++i)
      srow[(size_t)((b * 16 + h) * 1024) + q0 + i + lh * 8] = s[i];
  }
  // out_row = (sum_j t_j * V_j) / s   (attn_out layout: [b, s, h*64+d])
#pragma unroll
  for (int nb = 0; nb < 4; ++nb) {
#pragma unroll
    for (int i = 0; i < 8; ++i) {
      int row = q0 + i + lh * 8;
      float inv = 1.f / s[i];
      attn_out[(size_t)(b * 1024 + row) * 1024 + h * 64 + nb * 16 + l15] =
          acc[nb][i] * inv;
    }
  }
}

// ---------------------------------------------------------------------------
// Renormalize filtered_attn rows by their mask sums (streaming float4 pass).
// ---------------------------------------------------------------------------
__global__ __launch_bounds__(256) void rescale_kernel(
    float* __restrict__ fa, const float* __restrict__ srow) {
  size_t idx = (size_t)blockIdx.x * blockDim.x + threadIdx.x;  // float4 index
  size_t row = idx >> 8;  // 256 float4 per 1024-wide row
  float inv = 1.f / srow[row];
  float4* p = reinterpret_cast<float4*>(fa) + idx;
  float4 v = *p;
  v.x *= inv; v.y *= inv; v.z *= inv; v.w *= inv;
  *p = v;
}

// ---------------------------------------------------------------------------
extern "C" void kernel_launch(void* const* d_in, const int* in_sizes, int n_in,
                              void* d_out, int out_size, void* d_ws,
                              size_t ws_size, hipStream_t stream) {
  const float* x      = (const float*)d_in[0];  // [4,1024,1024]
  const float* w_qkv  = (const float*)d_in[1];  // [1024,3072]
  const float* b_qkv  = (const float*)d_in[2];  // [3072]
  const float* w_proj = (const float*)d_in[3];  // [1024,1024]
  const float* b_proj = (const float*)d_in[4];  // [1024]

  float* out = (float*)d_out;                    // [4096][1024]
  float* fa  = out + (size_t)4096 * 1024;        // [65536][1024] filtered_attn

  float* qkv      = (float*)d_ws;                      // [4096][3072]  50.3 MB
  float* attn_out = qkv + (size_t)4096 * 3072;         // [4096][1024]  16.8 MB
  float* srow     = attn_out + (size_t)4096 * 1024;    // [65536]        0.3 MB

  dim3 blk(256);
  // 1) QKV projection: [4096,1024] @ [1024,3072] + bias
  gemm_f32_wmma<<<dim3(3072 / 128, 4096 / 64), blk, 0, stream>>>(
      x, w_qkv, b_qkv, qkv, 4096, 3072, 1024);
  // 2) Threshold attention (writes unnormalized fa, srow, attn_out)
  attn_kernel<<<dim3(8, 16, 4), blk, 0, stream>>>(qkv, fa, attn_out, srow);
  // 3) Renormalize filtered_attn in place
  rescale_kernel<<<dim3(65536), blk, 0, stream>>>(fa, srow);
  // 4) Output projection: [4096,1024] @ [1024,1024] + bias
  gemm_f32_wmma<<<dim3(1024 / 128, 4096 / 64), blk, 0, stream>>>(
      attn_out, w_proj, b_proj, out, 4096, 1024, 1024);
}